// FantasticFour_45122926411974
// MI455X (gfx1250) — compile-verified
//
#include <hip/hip_runtime.h>
#include <math.h>

// ---------------------------------------------------------------------------
// Spectral-norm power iteration (4 chains) on K (2048 x 2048 x 3 x 3 f32).
// All heavy contractions are fused into two skinny GEMMs per iteration:
//   row pass:  D[o, 0..15]  = sum_c  K[o, c] * B[c, n]      (v3 | v1 | v2 | u4)
//   col pass:  D[i, 0..15]  = sum_r  K[o(r), i*9+hw(r)] * W[r, n]  (u1|u2|u3|v4)
// implemented with V_WMMA_F32_16X16X4_F32 (exact f32 math).
// K (151 MB) fits in the 192 MB L2, so the ~10 full passes are L2-resident.
// ---------------------------------------------------------------------------

typedef float v2f __attribute__((ext_vector_type(2)));
typedef float v8f __attribute__((ext_vector_type(8)));

#define O_DIM 2048
#define I_DIM 2048
#define CK    18432          // I*9 == O*9
#define NTOT  37748736       // 2048*2048*9
#define SLICES 16
#define CS     1152          // CK / SLICES  (multiple of 4)
#define PART_STRIDE 32768    // 2048 * 16

// ------------------------- small-vector L2 normalize -----------------------

__global__ __launch_bounds__(256) void sn_l2copy4(
    const float* s0, float* d0, int n0, const float* s1, float* d1, int n1,
    const float* s2, float* d2, int n2, const float* s3, float* d3, int n3) {
  const float* s; float* d; int n;
  switch (blockIdx.x) {
    case 0:  s = s0; d = d0; n = n0; break;
    case 1:  s = s1; d = d1; n = n1; break;
    case 2:  s = s2; d = d2; n = n2; break;
    default: s = s3; d = d3; n = n3; break;
  }
  if (s == nullptr || n <= 0) return;
  __shared__ float red[256];
  __shared__ float sc;
  float loc = 0.f;
  for (int i = threadIdx.x; i < n; i += 256) { float x = s[i]; loc += x * x; }
  red[threadIdx.x] = loc; __syncthreads();
  for (int st = 128; st > 0; st >>= 1) {
    if ((int)threadIdx.x < st) red[threadIdx.x] += red[threadIdx.x + st];
    __syncthreads();
  }
  if (threadIdx.x == 0) sc = 1.0f / (sqrtf(red[0]) + 1e-12f);
  __syncthreads();
  float m = sc;
  for (int i = threadIdx.x; i < n; i += 256) d[i] = s[i] * m;
}

__global__ __launch_bounds__(256) void sn_l2norm4(
    float* p0, int n0, float* p1, int n1, float* p2, int n2, float* p3, int n3) {
  float* p; int n;
  switch (blockIdx.x) {
    case 0:  p = p0; n = n0; break;
    case 1:  p = p1; n = n1; break;
    case 2:  p = p2; n = n2; break;
    default: p = p3; n = n3; break;
  }
  if (p == nullptr || n <= 0) return;
  __shared__ float red[256];
  __shared__ float sc;
  float loc = 0.f;
  for (int i = threadIdx.x; i < n; i += 256) { float x = p[i]; loc += x * x; }
  red[threadIdx.x] = loc; __syncthreads();
  for (int st = 128; st > 0; st >>= 1) {
    if ((int)threadIdx.x < st) red[threadIdx.x] += red[threadIdx.x + st];
    __syncthreads();
  }
  if (threadIdx.x == 0) sc = 1.0f / (sqrtf(red[0]) + 1e-12f);
  __syncthreads();
  float m = sc;
  for (int i = threadIdx.x; i < n; i += 256) p[i] *= m;
}

__global__ __launch_bounds__(1024) void sn_zero(float* p, int n) {
  int i = blockIdx.x * 1024 + threadIdx.x;
  if (i < n) p[i] = 0.f;
}

// ------------------------- B / W matrix builders ---------------------------
// Row-pass B[c][n], c = i*9 + h*3 + w:
//   n=0      : u3[c]
//   n=1..3   : (h == n-1) ? u1[i*3+w] : 0
//   n=4..6   : (w == n-4) ? u2[i*3+h] : 0
//   n=7..15  : (hw == n-7) ? v4[i]    : 0
__global__ __launch_bounds__(1024) void sn_build_row_B(
    const float* __restrict__ u1, const float* __restrict__ u2,
    const float* __restrict__ u3, const float* __restrict__ v4,
    float* __restrict__ B) {
  int idx = blockIdx.x * 1024 + threadIdx.x;
  if (idx >= CK * 16) return;
  int c = idx >> 4, n = idx & 15;
  int i = c / 9, hw = c - i * 9, h = hw / 3, w = hw - h * 3;
  float val;
  if (n == 0)       val = u3[c];
  else if (n <= 3)  val = (h == n - 1) ? u1[i * 3 + w] : 0.f;
  else if (n <= 6)  val = (w == n - 4) ? u2[i * 3 + h] : 0.f;
  else              val = (hw == n - 7) ? v4[i] : 0.f;
  B[idx] = val;
}

// Col-pass W[r][n], r = o*9 + h*3 + w:
//   n=0..2   : (w == n)   ? v1[o*3+h] : 0
//   n=3..5   : (h == n-3) ? v2[o*3+w] : 0
//   n=6..14  : (hw == n-6)? v3[o]     : 0
//   n=15     : u4[r]
__global__ __launch_bounds__(1024) void sn_build_col_W(
    const float* __restrict__ v1, const float* __restrict__ v2,
    const float* __restrict__ v3, const float* __restrict__ u4,
    float* __restrict__ W) {
  int idx = blockIdx.x * 1024 + threadIdx.x;
  if (idx >= CK * 16) return;
  int r = idx >> 4, n = idx & 15;
  int o = r / 9, hw = r - o * 9, h = hw / 3, w = hw - h * 3;
  float val;
  if (n <= 2)       val = (w == n) ? v1[o * 3 + h] : 0.f;
  else if (n <= 5)  val = (h == n - 3) ? v2[o * 3 + w] : 0.f;
  else if (n <= 14) val = (hw == n - 6) ? v3[o] : 0.f;
  else              val = u4[r];
  W[idx] = val;
}

// ------------------------- WMMA skinny GEMMs -------------------------------
// 2048 waves = 128 o/i-tiles x 16 k-slices; partials[slice][2048][16].
__global__ __launch_bounds__(256) void sn_gemm_row(
    const float* __restrict__ K, const float* __restrict__ B,
    float* __restrict__ part) {
  const int lane  = threadIdx.x & 31;
  const int wid   = (blockIdx.x * 256 + threadIdx.x) >> 5;
  const int tile  = wid >> 4;        // 0..127 -> 16 rows of K
  const int slice = wid & 15;        // k-range slice
  const int kh    = lane >> 4;       // lane-half selects K pair {0,1} or {2,3}
  const int mn    = lane & 15;       // A: M index, B: N index
  const float* Arow = K + (size_t)(tile * 16 + mn) * CK;
  const int cb = slice * CS;
  v8f acc = {0.f, 0.f, 0.f, 0.f, 0.f, 0.f, 0.f, 0.f};
  for (int c0 = cb; c0 < cb + CS; c0 += 4) {
    const int ka = c0 + 2 * kh;
    v2f a = *(const v2f*)(Arow + ka);          // A[m, ka], A[m, ka+1]
    v2f b;
    b.x = B[ka * 16 + mn];                     // B[ka,   n]
    b.y = B[(ka + 1) * 16 + mn];               // B[ka+1, n]
    acc = __builtin_amdgcn_wmma_f32_16x16x4_f32(
        false, a, false, b, (short)0, acc, false, false);
  }
  float* p = part + (size_t)slice * PART_STRIDE;
#pragma unroll
  for (int r = 0; r < 8; ++r)                  // D[r+8*kh, mn] in acc[r]
    p[(tile * 16 + r + 8 * kh) * 16 + mn] = acc[r];
}

__global__ __launch_bounds__(256) void sn_gemm_col(
    const float* __restrict__ K, const float* __restrict__ W,
    float* __restrict__ part) {
  const int lane  = threadIdx.x & 31;
  const int wid   = (blockIdx.x * 256 + threadIdx.x) >> 5;
  const int tile  = wid >> 4;        // 0..127 -> 16 columns (i values)
  const int slice = wid & 15;
  const int kh    = lane >> 4;
  const int mn    = lane & 15;
  const int ibase = tile * 16;
  const int rb = slice * CS;
  v8f acc = {0.f, 0.f, 0.f, 0.f, 0.f, 0.f, 0.f, 0.f};
  for (int r0 = rb; r0 < rb + CS; r0 += 4) {
    const int k0 = r0 + 2 * kh;
    const int k1 = k0 + 1;
    const int o0 = k0 / 9, hw0 = k0 - o0 * 9;
    const int o1 = k1 / 9, hw1 = k1 - o1 * 9;
    v2f a;
    a.x = K[(size_t)o0 * CK + (ibase + mn) * 9 + hw0];  // K2[k0, i]
    a.y = K[(size_t)o1 * CK + (ibase + mn) * 9 + hw1];  // K2[k1, i]
    v2f b;
    b.x = W[k0 * 16 + mn];
    b.y = W[k1 * 16 + mn];
    acc = __builtin_amdgcn_wmma_f32_16x16x4_f32(
        false, a, false, b, (short)0, acc, false, false);
  }
  float* p = part + (size_t)slice * PART_STRIDE;
#pragma unroll
  for (int r = 0; r < 8; ++r)
    p[(ibase + r + 8 * kh) * 16 + mn] = acc[r];
}

// ------------------------- slice reductions --------------------------------

__global__ __launch_bounds__(1024) void sn_reduce_row(
    const float* __restrict__ part, float* v1, float* v2, float* v3, float* u4) {
  int idx = blockIdx.x * 1024 + threadIdx.x;
  if (idx >= PART_STRIDE) return;
  int o = idx >> 4, n = idx & 15;
  float s = 0.f;
  for (int sl = 0; sl < SLICES; ++sl) s += part[(size_t)sl * PART_STRIDE + idx];
  if (n == 0)      v3[o] = s;
  else if (n <= 3) v1[o * 3 + (n - 1)] = s;
  else if (n <= 6) v2[o * 3 + (n - 4)] = s;
  else             u4[o * 9 + (n - 7)] = s;
}

__global__ __launch_bounds__(1024) void sn_reduce_col(
    const float* __restrict__ part, float* u1, float* u2, float* u3, float* v4,
    int writeU, int writeV4) {
  int idx = blockIdx.x * 1024 + threadIdx.x;
  if (idx >= PART_STRIDE) return;
  int i = idx >> 4, n = idx & 15;
  float s = 0.f;
  for (int sl = 0; sl < SLICES; ++sl) s += part[(size_t)sl * PART_STRIDE + idx];
  if (n <= 2)       { if (writeU)  u1[i * 3 + n] = s; }
  else if (n <= 5)  { if (writeU)  u2[i * 3 + (n - 3)] = s; }
  else if (n <= 14) { if (writeU)  u3[i * 9 + (n - 6)] = s; }
  else              { if (writeV4) v4[i] = s; }
}

// ------------------------- sigma + final scale -----------------------------

__global__ __launch_bounds__(1024) void sn_sigma(
    const float* __restrict__ part, const float* __restrict__ v1,
    const float* __restrict__ v2, const float* __restrict__ v3,
    const float* __restrict__ u4, float* __restrict__ sg) {
  float s1 = 0.f, s2 = 0.f, s3 = 0.f, s4 = 0.f;
  for (int idx = threadIdx.x; idx < PART_STRIDE; idx += 1024) {
    int o = idx >> 4, n = idx & 15;
    float t = 0.f;
    for (int sl = 0; sl < SLICES; ++sl) t += part[(size_t)sl * PART_STRIDE + idx];
    if (n == 0)      s3 += t * v3[o];
    else if (n <= 3) s1 += t * v1[o * 3 + (n - 1)];
    else if (n <= 6) s2 += t * v2[o * 3 + (n - 4)];
    else             s4 += t * u4[o * 9 + (n - 7)];
  }
  __shared__ float r1[1024], r2[1024], r3[1024], r4[1024];
  r1[threadIdx.x] = s1; r2[threadIdx.x] = s2;
  r3[threadIdx.x] = s3; r4[threadIdx.x] = s4;
  __syncthreads();
  for (int st = 512; st > 0; st >>= 1) {
    if ((int)threadIdx.x < st) {
      r1[threadIdx.x] += r1[threadIdx.x + st];
      r2[threadIdx.x] += r2[threadIdx.x + st];
      r3[threadIdx.x] += r3[threadIdx.x + st];
      r4[threadIdx.x] += r4[threadIdx.x + st];
    }
    __syncthreads();
  }
  if (threadIdx.x == 0) {
    float sig = fminf(fminf(r1[0], r2[0]), fminf(r3[0], r4[0]));
    sg[0] = sig;
    sg[1] = 1.0f / sig;
  }
}

__global__ __launch_bounds__(1024) void sn_scale(
    const float* __restrict__ K, const float* __restrict__ sg,
    float* __restrict__ out) {
  size_t i = (size_t)blockIdx.x * 1024 + threadIdx.x;   // float4 index
  float inv = sg[1];
  if (i < (size_t)(NTOT / 4)) {
    float4 k = ((const float4*)K)[i];
    k.x *= inv; k.y *= inv; k.z *= inv; k.w *= inv;
    ((float4*)out)[i] = k;
  }
}

// ------------------------- host launch sequence ----------------------------

extern "C" void kernel_launch(void* const* d_in, const int* in_sizes, int n_in,
                              void* d_out, int out_size, void* d_ws, size_t ws_size,
                              hipStream_t stream) {
  (void)in_sizes; (void)n_in; (void)out_size; (void)ws_size;
  const float* K    = (const float*)d_in[0];
  const float* u1in = (const float*)d_in[1];
  const float* u2in = (const float*)d_in[2];
  const float* u3in = (const float*)d_in[3];
  const float* u4in = (const float*)d_in[4];
  float* out = (float*)d_out;

  float* ws = (float*)d_ws;
  float* U1 = ws;             // 6144
  float* U2 = U1 + 6144;      // 6144
  float* U3 = U2 + 6144;      // 18432
  float* U4 = U3 + 18432;     // 18432
  float* V1 = U4 + 18432;     // 6144   (V1,V2,V3 contiguous for zero-fill)
  float* V2 = V1 + 6144;      // 6144
  float* V3 = V2 + 6144;      // 2048
  float* V4 = V3 + 2048;      // 2048
  float* BM = V4 + 2048;      // 294912 (18432 x 16)
  float* PT = BM + 294912;    // 524288 (16 x 2048 x 16)
  float* SG = PT + 524288;    // 8

  // u = l2n(u_in) for all four chains (do not mutate inputs)
  sn_l2copy4<<<4, 256, 0, stream>>>(u1in, U1, 6144, u2in, U2, 6144,
                                    u3in, U3, 18432, u4in, U4, 18432);
  // warm-up col pass computes v4^1 only (v1,v2,v3 zeroed so junk u-cols skipped)
  sn_zero<<<14, 1024, 0, stream>>>(V1, 14336);
  sn_build_col_W<<<288, 1024, 0, stream>>>(V1, V2, V3, U4, BM);
  sn_gemm_col<<<256, 256, 0, stream>>>(K, BM, PT);
  sn_reduce_col<<<32, 1024, 0, stream>>>(PT, U1, U2, U3, V4, 0, 1);
  sn_l2norm4<<<1, 256, 0, stream>>>(V4, 2048, nullptr, 0, nullptr, 0, nullptr, 0);

  for (int t = 0; t < 3; ++t) {
    // row pass: v1,v2,v3 (from u1,u2,u3) and u4 (from current v4)
    sn_build_row_B<<<288, 1024, 0, stream>>>(U1, U2, U3, V4, BM);
    sn_gemm_row<<<256, 256, 0, stream>>>(K, BM, PT);
    sn_reduce_row<<<32, 1024, 0, stream>>>(PT, V1, V2, V3, U4);
    sn_l2norm4<<<4, 256, 0, stream>>>(V1, 6144, V2, 6144, V3, 2048, U4, 18432);
    // col pass: u1,u2,u3 (from v1,v2,v3) and next v4 (from u4) unless last iter
    sn_build_col_W<<<288, 1024, 0, stream>>>(V1, V2, V3, U4, BM);
    sn_gemm_col<<<256, 256, 0, stream>>>(K, BM, PT);
    const int wv4 = (t < 2) ? 1 : 0;
    sn_reduce_col<<<32, 1024, 0, stream>>>(PT, U1, U2, U3, V4, 1, wv4);
    if (wv4)
      sn_l2norm4<<<4, 256, 0, stream>>>(U1, 6144, U2, 6144, U3, 18432, V4, 2048);
    else
      sn_l2norm4<<<3, 256, 0, stream>>>(U1, 6144, U2, 6144, U3, 18432, nullptr, 0);
  }

  // sigma pass: one more row GEMM with final u's, weighted by final v's
  sn_build_row_B<<<288, 1024, 0, stream>>>(U1, U2, U3, V4, BM);
  sn_gemm_row<<<256, 256, 0, stream>>>(K, BM, PT);
  sn_sigma<<<1, 1024, 0, stream>>>(PT, V1, V2, V3, U4, SG);
  // out = K / sigma
  sn_scale<<<9216, 1024, 0, stream>>>(K, SG, out);
}